// TemporalGNN_35459249996211
// MI455X (gfx1250) — compile-verified
//
#include <hip/hip_runtime.h>
#include <hip/hip_bf16.h>

#define Nn 20000
#define Ee 640000
#define Ff 128
#define Hh 128

typedef __attribute__((ext_vector_type(2))) float v2f;
typedef __attribute__((ext_vector_type(8))) float v8f;

// ---------------- zero workspace accumulators ----------------
__global__ void k_zero(float* p, size_t n) {
    size_t i = (size_t)blockIdx.x * blockDim.x + threadIdx.x;
    if (i < n) p[i] = 0.0f;
}

// ---------------- degree accumulation ----------------
__global__ void k_deg(const int* __restrict__ src, const float* __restrict__ ew,
                      float* __restrict__ deg) {
    int e = blockIdx.x * blockDim.x + threadIdx.x;
    if (e < Ee) atomicAdd(&deg[src[e]], ew[e]);
}

// ---------------- dis = deg>0 ? rsqrt(max(deg,1e-12)) : 0  (in place) ----------------
__global__ void k_dis(float* __restrict__ deg) {
    int i = blockIdx.x * blockDim.x + threadIdx.x;
    if (i < Nn) {
        float d = deg[i];
        deg[i] = (d > 0.0f) ? rsqrtf(fmaxf(d, 1e-12f)) : 0.0f;
    }
}

// ---------------- edge scatter: Lx[dst] += -w_sym * x[src]; same for h ----------------
// one block (128 threads) per edge; feature index = threadIdx.x
__global__ void k_scatter(const float* __restrict__ x, const float* __restrict__ h,
                          const int* __restrict__ src, const int* __restrict__ dst,
                          const float* __restrict__ ew, const float* __restrict__ dis,
                          float* __restrict__ Lx, float* __restrict__ Lh) {
    int e = blockIdx.x;
    int f = threadIdx.x;                 // 0..127
    int s = src[e], d = dst[e];
    float w = -dis[s] * ew[e] * dis[d];  // minus sign of L_hat folded in
    atomicAdd(&Lx[(size_t)d * Ff + f], w * x[(size_t)s * Ff + f]);
    atomicAdd(&Lh[(size_t)d * Hh + f], w * h[(size_t)s * Hh + f]);
}

// ---------------- WMMA f32 GEMM accumulate: acc += A[row0:,128] * B[:,col0:] ----------------
// A is (.,128) row-major, B is (128,128) row-major (one gate/k slice of Wx/Wh).
__device__ __forceinline__ v8f wmma_acc(const float* __restrict__ A,
                                        const float* __restrict__ B,
                                        int row0, int col0, int lane, v8f acc) {
    int m  = lane & 15;           // M row (A) / N col (B) within tile
    int kb = (lane >> 4) << 1;    // lanes 16..31 hold K+2,K+3
    const float* arow = A + (size_t)(row0 + m) * 128;
    const float* bcol = B + col0 + m;
#pragma unroll
    for (int kk = 0; kk < 128; kk += 4) {
        v2f a, b;
        a.x = arow[kk + kb];
        a.y = arow[kk + kb + 1];
        b.x = bcol[(size_t)(kk + kb) * 128];
        b.y = bcol[(size_t)(kk + kb + 1) * 128];
        // D = A*B + C, exact f32; emits v_wmma_f32_16x16x4_f32
        acc = __builtin_amdgcn_wmma_f32_16x16x4_f32(
            false, a, false, b, (short)0, acc, false, false);
    }
    return acc;
}

// G(N x 512) = x*Wx[g,0] + Lx*Wx[g,1] + h*Wh[g,0] + Lh*Wh[g,1], gates concatenated
// 8 waves/block, one 16x16 output tile per wave. tiles = 1250 * 32 = 40000 -> 5000 blocks.
__global__ void k_gemm(const float* __restrict__ x,  const float* __restrict__ Lx,
                       const float* __restrict__ h,  const float* __restrict__ Lh,
                       const float* __restrict__ Wx, const float* __restrict__ Wh,
                       float* __restrict__ G) {
    int wave = threadIdx.x >> 5;
    int lane = threadIdx.x & 31;
    int tile = blockIdx.x * 8 + wave;   // < 40000
    int rt = tile >> 5;                 // row tile: 0..1249
    int ct = tile & 31;                 // col tile: 0..31
    int row0 = rt * 16;
    int g  = ct >> 3;                   // gate 0..3
    int cg = (ct & 7) * 16;             // column within gate's 128

    const size_t WSLICE = 128 * 128;

    v8f acc = {};
    acc = wmma_acc(x,  Wx + (size_t)(g * 2 + 0) * WSLICE, row0, cg, lane, acc);
    acc = wmma_acc(Lx, Wx + (size_t)(g * 2 + 1) * WSLICE, row0, cg, lane, acc);
    acc = wmma_acc(h,  Wh + (size_t)(g * 2 + 0) * WSLICE, row0, cg, lane, acc);
    acc = wmma_acc(Lh, Wh + (size_t)(g * 2 + 1) * WSLICE, row0, cg, lane, acc);

    // C/D layout: VGPR v -> M = v (lanes 0..15) or v+8 (lanes 16..31); N = lane&15
    int n  = lane & 15;
    int mb = (lane >> 4) * 8;
    int col = g * 128 + cg + n;
#pragma unroll
    for (int v = 0; v < 8; ++v) {
        G[(size_t)(row0 + mb + v) * 512 + col] = acc[v];
    }
}

// ---------------- LSTM gate pointwise ----------------
__device__ __forceinline__ float sigf(float x) { return 1.0f / (1.0f + __expf(-x)); }

__global__ void k_gates(const float* __restrict__ G, const float* __restrict__ c,
                        const float* __restrict__ b, const float* __restrict__ wc,
                        float* __restrict__ out) {
    size_t idx = (size_t)blockIdx.x * blockDim.x + threadIdx.x;
    if (idx >= (size_t)Nn * Hh) return;
    size_t node = idx >> 7;       // /128
    int j = (int)(idx & 127);
    const float* g = G + node * 512;
    float cv = c[idx];

    float i_g = sigf(g[j]        + wc[j]       * cv + b[j]);
    float f_g = sigf(g[128 + j]  + wc[128 + j] * cv + b[128 + j]);
    float g_g = tanhf(g[256 + j] + b[256 + j]);
    float cn  = f_g * cv + i_g * g_g;
    float o_g = sigf(g[384 + j]  + wc[256 + j] * cn + b[384 + j]);
    float hn  = o_g * tanhf(cn);

    out[idx] = hn;                              // out (== h_new)
    out[(size_t)Nn * Hh + idx] = hn;            // h_new
    out[2 * (size_t)Nn * Hh + idx] = cn;        // c_new
}

extern "C" void kernel_launch(void* const* d_in, const int* in_sizes, int n_in,
                              void* d_out, int out_size, void* d_ws, size_t ws_size,
                              hipStream_t stream) {
    const float* x  = (const float*)d_in[0];   // (N,128)
    const float* ew = (const float*)d_in[1];   // (E,)
    const float* h  = (const float*)d_in[2];   // (N,128)
    const float* c  = (const float*)d_in[3];   // (N,128)
    const float* Wx = (const float*)d_in[4];   // (4,2,128,128)
    const float* Wh = (const float*)d_in[5];   // (4,2,128,128)
    const float* b  = (const float*)d_in[6];   // (4,128)
    const float* wc = (const float*)d_in[7];   // (3,128)
    const int*  src = (const int*)d_in[8];     // (E,)
    const int*  dst = (const int*)d_in[9];     // (E,)
    float* out = (float*)d_out;                // 3 * N * 128

    // workspace layout (floats)
    float* ws  = (float*)d_ws;
    float* deg = ws;                           // N   (becomes dis in-place)
    float* Lx  = deg + Nn;                     // N*128
    float* Lh  = Lx + (size_t)Nn * Ff;         // N*128
    float* G   = Lh + (size_t)Nn * Hh;         // N*512

    // 1) zero accumulators (deg + Lx + Lh) -- required every call (graph replay)
    size_t zcount = (size_t)Nn * (1 + Ff + Hh);
    k_zero<<<(unsigned)((zcount + 255) / 256), 256, 0, stream>>>(deg, zcount);

    // 2) degree
    k_deg<<<(Ee + 255) / 256, 256, 0, stream>>>(src, ew, deg);

    // 3) dis (in place)
    k_dis<<<(Nn + 255) / 256, 256, 0, stream>>>(deg);

    // 4) graph propagation for x and h
    k_scatter<<<Ee, 128, 0, stream>>>(x, h, src, dst, ew, deg, Lx, Lh);

    // 5) WMMA GEMM: G = x*Wx0 + Lx*Wx1 + h*Wh0 + Lh*Wh1 (per gate)
    k_gemm<<<5000, 256, 0, stream>>>(x, Lx, h, Lh, Wx, Wh, G);

    // 6) gates + outputs
    k_gates<<<(unsigned)(((size_t)Nn * Hh + 255) / 256), 256, 0, stream>>>(G, c, b, wc, out);
}